// GraphConv_8916352107017
// MI455X (gfx1250) — compile-verified
//
#include <hip/hip_runtime.h>
#include <hip/hip_bf16.h>

typedef __attribute__((ext_vector_type(2))) float v2f;
typedef __attribute__((ext_vector_type(8))) float v8f;

#define IN_F 256
#define OUT_F 256

// ---------------------------------------------------------------------------
// Kernel 1: edge scatter.  One wave32 per edge.
//   S[src, :] += val * X[dst, :]
// Each lane owns 8 contiguous floats of the 256-wide row (2x float4 loads),
// then 8 native f32 global atomics (non-returning -> STOREcnt path).
// X and S both live in the 192MB L2, so the gather + atomic traffic is
// L2-resident.
// ---------------------------------------------------------------------------
__global__ void gc_scatter_kernel(const float* __restrict__ X,
                                  const int* __restrict__ edge_src,
                                  const int* __restrict__ edge_dst,
                                  const float* __restrict__ edge_val,
                                  float* __restrict__ S,
                                  int n_edges) {
    int gtid = blockIdx.x * blockDim.x + threadIdx.x;
    int wave = gtid >> 5;
    int lane = threadIdx.x & 31;
    if (wave >= n_edges) return;

    int src = edge_src[wave];
    int dst = edge_dst[wave];
    float v = edge_val[wave];

    const float4* xrow = (const float4*)(X + (size_t)dst * IN_F);
    float4 a = xrow[lane * 2 + 0];
    float4 b = xrow[lane * 2 + 1];

    float* srow = S + (size_t)src * IN_F + lane * 8;
    unsafeAtomicAdd(srow + 0, a.x * v);
    unsafeAtomicAdd(srow + 1, a.y * v);
    unsafeAtomicAdd(srow + 2, a.z * v);
    unsafeAtomicAdd(srow + 3, a.w * v);
    unsafeAtomicAdd(srow + 4, b.x * v);
    unsafeAtomicAdd(srow + 5, b.y * v);
    unsafeAtomicAdd(srow + 6, b.z * v);
    unsafeAtomicAdd(srow + 7, b.w * v);
}

// ---------------------------------------------------------------------------
// Kernel 2: out = S @ W^T + b  via V_WMMA_F32_16X16X4_F32.
// One wave computes a 16x32 output tile (two accumulators share the A frag).
//   out[m,n] = sum_k S[m,k] * W[n,k]   -> B column n == W row n (contiguous).
// A frag (16x4 f32, 2 VGPR): lane L: m = L&15, k-pair base = 2*(L>>4)
// B frag (4x16 f32, 2 VGPR): lane L: n = L&15, k-pair base = 2*(L>>4)
// C/D (16x16 f32, 8 VGPR):   lane L: n = L&15, vgpr i -> m = i + 8*(L>>4)
// M = 100000 = 6250 * 16 exactly -> no tail handling.
// ---------------------------------------------------------------------------
__global__ void gc_gemm_kernel(const float* __restrict__ S,
                               const float* __restrict__ W,
                               const float* __restrict__ bias,
                               float* __restrict__ out,
                               int M) {
    const int NT = OUT_F / 32;                     // 8 n-tiles of width 32
    int gtid = blockIdx.x * blockDim.x + threadIdx.x;
    int wave = gtid >> 5;
    int lane = threadIdx.x & 31;
    int mt = wave / NT;
    int nt = wave % NT;
    if (mt * 16 >= M) return;

    int half = lane >> 4;   // 0 or 1
    int l15  = lane & 15;

    const float* arow  = S + (size_t)(mt * 16 + l15) * IN_F + 2 * half;
    const float* brow0 = W + (size_t)(nt * 32 + l15) * IN_F + 2 * half;
    const float* brow1 = W + (size_t)(nt * 32 + 16 + l15) * IN_F + 2 * half;

    v8f c0 = {};
    v8f c1 = {};

    #pragma unroll 4
    for (int k = 0; k < IN_F; k += 4) {
        v2f a  = *(const v2f*)(arow + k);
        v2f b0 = *(const v2f*)(brow0 + k);
        v2f b1 = *(const v2f*)(brow1 + k);
        c0 = __builtin_amdgcn_wmma_f32_16x16x4_f32(
                 false, a, false, b0, (short)0, c0, false, false);
        c1 = __builtin_amdgcn_wmma_f32_16x16x4_f32(
                 false, a, false, b1, (short)0, c1, false, false);
    }

    float bias0 = bias[nt * 32 + l15];
    float bias1 = bias[nt * 32 + 16 + l15];

    float* orow = out + (size_t)(mt * 16 + 8 * half) * OUT_F + nt * 32;
    #pragma unroll
    for (int i = 0; i < 8; ++i) {
        orow[(size_t)i * OUT_F + l15]      = c0[i] + bias0;
        orow[(size_t)i * OUT_F + 16 + l15] = c1[i] + bias1;
    }
}

extern "C" void kernel_launch(void* const* d_in, const int* in_sizes, int n_in,
                              void* d_out, int out_size, void* d_ws, size_t ws_size,
                              hipStream_t stream) {
    const float* X        = (const float*)d_in[0];
    const int*   edge_src = (const int*)d_in[1];
    const int*   edge_dst = (const int*)d_in[2];
    const float* edge_val = (const float*)d_in[3];
    const float* W        = (const float*)d_in[4];
    const float* bias     = (const float*)d_in[5];

    int n_nodes = in_sizes[0] / IN_F;     // 100000
    int n_edges = in_sizes[1];            // 3200000

    float* S = (float*)d_ws;              // [n_nodes, IN_F] accumulator
    size_t s_bytes = (size_t)n_nodes * IN_F * sizeof(float);

    // Zero accumulator (must happen every call; harness does not re-poison).
    hipMemsetAsync(S, 0, s_bytes, stream);

    // Scatter: one wave per edge, 8 waves per 256-thread block.
    {
        long long total_threads = (long long)n_edges * 32;
        int block = 256;
        int grid = (int)((total_threads + block - 1) / block);
        gc_scatter_kernel<<<grid, block, 0, stream>>>(X, edge_src, edge_dst,
                                                      edge_val, S, n_edges);
    }

    // GEMM: one wave per 16x32 tile; 8 n-tiles, n_nodes/16 m-tiles.
    {
        int mtiles = n_nodes / 16;                 // 6250
        long long waves = (long long)mtiles * (OUT_F / 32);
        long long total_threads = waves * 32;
        int block = 256;
        int grid = (int)((total_threads + block - 1) / block);
        gc_gemm_kernel<<<grid, block, 0, stream>>>(S, W, bias, (float*)d_out,
                                                   n_nodes);
    }
}